// GraphPropagation_53068615910201
// MI455X (gfx1250) — compile-verified
//
#include <hip/hip_runtime.h>
#include <hip/hip_bf16.h>
#include <math.h>

// GraphPropagation on MI455X (gfx1250), fp32 end-to-end via V_WMMA_F32_16X16X4_F32.
// N=768 prototypes, D=128 feat, H=8 edge-MLP hidden, A=312 attributes.

typedef __attribute__((ext_vector_type(2))) float v2f;
typedef __attribute__((ext_vector_type(8))) float v8f;

#define NP   768
#define DF   128
#define HH   8
#define AD   312
#define EPS_ 1e-8f

static __device__ __forceinline__ v8f wmma_f32(v2f a, v2f b, v8f c) {
  return __builtin_amdgcn_wmma_f32_16x16x4_f32(
      /*neg_a=*/false, a, /*neg_b=*/false, b,
      /*c_mod=*/(short)0, c, /*reuse_a=*/false, /*reuse_b=*/false);
}

// ---------------- row L2 norms ----------------
__global__ void rownorm_kernel(const float* __restrict__ X, int K, float* __restrict__ out) {
  int row = blockIdx.x;
  __shared__ float red[256];
  float s = 0.f;
  for (int c = threadIdx.x; c < K; c += 256) {
    float v = X[row * K + c];
    s += v * v;
  }
  red[threadIdx.x] = s;
  __syncthreads();
  for (int st = 128; st > 0; st >>= 1) {
    if (threadIdx.x < (unsigned)st) red[threadIdx.x] += red[threadIdx.x + st];
    __syncthreads();
  }
  if (threadIdx.x == 0) out[row] = sqrtf(red[0]);
}

// ---------------- cosine-sim logits:  out = (X @ X^T) / max(n_i n_j, eps) * invtemp ----------------
__global__ void gram_logits_kernel(const float* __restrict__ X, const float* __restrict__ nrm,
                                   int K, float invtemp, float* __restrict__ out) {
  int tj = blockIdx.x, ti = blockIdx.y;
  int lane = threadIdx.x;
  int ml = lane & 15;
  int koff = (lane >> 4) * 2;
  const float* arow = X + (size_t)(ti * 16 + ml) * K;
  const float* brow = X + (size_t)(tj * 16 + ml) * K;
  v8f acc = {};
  for (int k = 0; k < K; k += 4) {
    int k0 = k + koff;
    v2f a = {arow[k0], arow[k0 + 1]};
    v2f b = {brow[k0], brow[k0 + 1]};
    acc = wmma_f32(a, b, acc);
  }
  int mbase = (lane >> 4) * 8;
  int j = tj * 16 + ml;
  float nj = nrm[j];
#pragma unroll
  for (int r = 0; r < 8; r++) {
    int i = ti * 16 + mbase + r;
    float denom = fmaxf(nrm[i] * nj, EPS_);
    out[(size_t)i * NP + j] = acc[r] / denom * invtemp;
  }
}

// ---------------- generic C = A @ B  (A: MxK row-major, B: KxN row-major) ----------------
__global__ void gemm_ab_kernel(const float* __restrict__ A, const float* __restrict__ B,
                               int K, int lda, int ldb, float* __restrict__ C, int ldc) {
  int tj = blockIdx.x, ti = blockIdx.y;
  int lane = threadIdx.x;
  int ml = lane & 15;
  int koff = (lane >> 4) * 2;
  const float* arow = A + (size_t)(ti * 16 + ml) * lda;
  int bcol = tj * 16 + ml;
  v8f acc = {};
  for (int k = 0; k < K; k += 4) {
    int k0 = k + koff;
    v2f a = {arow[k0], arow[k0 + 1]};
    v2f b = {B[(size_t)k0 * ldb + bcol], B[(size_t)(k0 + 1) * ldb + bcol]};
    acc = wmma_f32(a, b, acc);
  }
  int mbase = (lane >> 4) * 8;
  int j = tj * 16 + ml;
#pragma unroll
  for (int r = 0; r < 8; r++) {
    int i = ti * 16 + mbase + r;
    C[(size_t)i * ldc + j] = acc[r];
  }
}

// ---------------- t = (P + PE) @ W^T + bias   (M=768, N=K=128, W row-major [128x128]) ----------------
__global__ void gemm_xwT_bias_kernel(const float* __restrict__ P, const float* __restrict__ PE,
                                     const float* __restrict__ W, const float* __restrict__ bias,
                                     float* __restrict__ T) {
  int tj = blockIdx.x, ti = blockIdx.y;
  int lane = threadIdx.x;
  int ml = lane & 15;
  int koff = (lane >> 4) * 2;
  size_t arow0 = (size_t)(ti * 16 + ml) * DF;
  size_t brow0 = (size_t)(tj * 16 + ml) * DF;  // B[k][n] = W[n][k]
  v8f acc = {};
  for (int k = 0; k < DF; k += 4) {
    int k0 = k + koff;
    v2f a = {P[arow0 + k0] + PE[arow0 + k0], P[arow0 + k0 + 1] + PE[arow0 + k0 + 1]};
    v2f b = {W[brow0 + k0], W[brow0 + k0 + 1]};
    acc = wmma_f32(a, b, acc);
  }
  int mbase = (lane >> 4) * 8;
  int j = tj * 16 + ml;
  float bj = bias[j];
#pragma unroll
  for (int r = 0; r < 8; r++) {
    int i = ti * 16 + mbase + r;
    T[(size_t)i * DF + j] = acc[r] + bj;
  }
}

// ---------------- out = relu(LayerNorm(t)*g + beta) + p   (rows of 128) ----------------
__global__ void ln_relu_add_kernel(const float* __restrict__ T, const float* __restrict__ P,
                                   const float* __restrict__ g, const float* __restrict__ beta,
                                   float* __restrict__ out) {
  int row = blockIdx.x;
  int c = threadIdx.x;
  __shared__ float red[128];
  float x = T[(size_t)row * DF + c];
  red[c] = x;
  __syncthreads();
  for (int st = 64; st > 0; st >>= 1) {
    if (c < st) red[c] += red[c + st];
    __syncthreads();
  }
  float mean = red[0] * (1.f / DF);
  __syncthreads();
  float d = x - mean;
  red[c] = d * d;
  __syncthreads();
  for (int st = 64; st > 0; st >>= 1) {
    if (c < st) red[c] += red[c + st];
    __syncthreads();
  }
  float var = red[0] * (1.f / DF);
  float y = d * rsqrtf(var + 1e-5f) * g[c] + beta[c];
  out[(size_t)row * DF + c] = fmaxf(y, 0.f) + P[(size_t)row * DF + c];
}

// ---------------- q[i,h] = sum_k W1[h,k] * x[i,k]^2 ----------------
__global__ void q_kernel(const float* __restrict__ X, const float* __restrict__ W1,
                         float* __restrict__ q) {
  int id = blockIdx.x * blockDim.x + threadIdx.x;
  if (id >= NP * HH) return;
  int i = id >> 3, h = id & 7;
  float s = 0.f;
  for (int k = 0; k < DF; k++) {
    float v = X[(size_t)i * DF + k];
    s += W1[h * DF + k] * v * v;
  }
  q[id] = s;
}

// ---------------- fused pairwise edge MLP tile ----------------
// y_h(i,j) = q[i,h] + q[j,h] + b1[h] - 2 * sum_k (W1[h,k]*x[i,k]) * x[j,k]
// e(i,j)   = tanh( W2 . relu(instnorm_H(y)) + b2 )
// out      = e * (last + EPS) * invtemp     (row-softmaxed by a later kernel)
__global__ void edge_tile_kernel(const float* __restrict__ X, const float* __restrict__ q,
                                 const float* __restrict__ W1g, const float* __restrict__ b1g,
                                 const float* __restrict__ W2g, const float* __restrict__ b2g,
                                 const float* __restrict__ last, float invtemp,
                                 float* __restrict__ out) {
  __shared__ float sW1[HH * DF];
  __shared__ float sW2[HH];
  __shared__ float sb1[HH];
  int lane = threadIdx.x;
  for (int idx = lane; idx < HH * DF; idx += 32) sW1[idx] = W1g[idx];
  if (lane < HH) {
    sW2[lane] = W2g[lane];
    sb1[lane] = b1g[lane];
  }
  __syncthreads();

  int tj = blockIdx.x, ti = blockIdx.y;
  int ml = lane & 15;
  int koff = (lane >> 4) * 2;
  const float* arow = X + (size_t)(ti * 16 + ml) * DF;
  const float* brow = X + (size_t)(tj * 16 + ml) * DF;

  v8f acc[HH] = {};
  for (int k = 0; k < DF; k += 4) {
    int k0 = k + koff;
    float a0 = arow[k0], a1 = arow[k0 + 1];
    v2f b = {brow[k0], brow[k0 + 1]};
#pragma unroll
    for (int h = 0; h < HH; h++) {
      v2f a = {a0 * sW1[h * DF + k0], a1 * sW1[h * DF + k0 + 1]};
      acc[h] = wmma_f32(a, b, acc[h]);
    }
  }

  float b2 = b2g[0];
  int mbase = (lane >> 4) * 8;
  int j = tj * 16 + ml;
#pragma unroll
  for (int r = 0; r < 8; r++) {
    int i = ti * 16 + mbase + r;
    float y[HH];
    float mean = 0.f;
#pragma unroll
    for (int h = 0; h < HH; h++) {
      y[h] = q[i * HH + h] + q[j * HH + h] + sb1[h] - 2.f * acc[h][r];
      mean += y[h];
    }
    mean *= (1.f / HH);
    float var = 0.f;
#pragma unroll
    for (int h = 0; h < HH; h++) {
      float d = y[h] - mean;
      var += d * d;
    }
    var *= (1.f / HH);
    float rstd = rsqrtf(var + 1e-5f);
    float s = 0.f;
#pragma unroll
    for (int h = 0; h < HH; h++) {
      float z = fmaxf((y[h] - mean) * rstd, 0.f);
      s += z * sW2[h];
    }
    float e = tanhf(s + b2);
    out[(size_t)i * NP + j] = e * (last[(size_t)i * NP + j] + EPS_) * invtemp;
  }
}

// ---------------- row softmax over 768 cols ----------------
__global__ void softmax768_kernel(const float* __restrict__ in, float* __restrict__ out) {
  int row = blockIdx.x;
  __shared__ float red[256];
  const float* r = in + (size_t)row * NP;
  float m = -INFINITY;
  for (int c = threadIdx.x; c < NP; c += 256) m = fmaxf(m, r[c]);
  red[threadIdx.x] = m;
  __syncthreads();
  for (int st = 128; st > 0; st >>= 1) {
    if (threadIdx.x < (unsigned)st)
      red[threadIdx.x] = fmaxf(red[threadIdx.x], red[threadIdx.x + st]);
    __syncthreads();
  }
  m = red[0];
  __syncthreads();
  float vals[3];
  float sum = 0.f;
#pragma unroll
  for (int k = 0; k < 3; k++) {
    int c = threadIdx.x + k * 256;
    vals[k] = expf(r[c] - m);
    sum += vals[k];
  }
  red[threadIdx.x] = sum;
  __syncthreads();
  for (int st = 128; st > 0; st >>= 1) {
    if (threadIdx.x < (unsigned)st) red[threadIdx.x] += red[threadIdx.x + st];
    __syncthreads();
  }
  float inv = 1.f / red[0];
#pragma unroll
  for (int k = 0; k < 3; k++) {
    int c = threadIdx.x + k * 256;
    out[(size_t)row * NP + c] = vals[k] * inv;
  }
}

extern "C" void kernel_launch(void* const* d_in, const int* in_sizes, int n_in,
                              void* d_out, int out_size, void* d_ws, size_t ws_size,
                              hipStream_t stream) {
  const float* VP   = (const float*)d_in[0];
  const float* SPm  = (const float*)d_in[1];
  const float* ATT  = (const float*)d_in[2];
  const float* Wvn  = (const float*)d_in[3];
  const float* bvn  = (const float*)d_in[4];
  const float* gvn  = (const float*)d_in[5];
  const float* betavn = (const float*)d_in[6];
  const float* Wve1 = (const float*)d_in[7];
  const float* bve1 = (const float*)d_in[8];
  const float* Wve2 = (const float*)d_in[9];
  const float* bve2 = (const float*)d_in[10];
  const float* Wsn  = (const float*)d_in[11];
  const float* bsn  = (const float*)d_in[12];
  const float* gsn  = (const float*)d_in[13];
  const float* betasn = (const float*)d_in[14];
  const float* Wse1 = (const float*)d_in[15];
  const float* bse1 = (const float*)d_in[16];
  const float* Wse2 = (const float*)d_in[17];
  const float* bse2 = (const float*)d_in[18];

  float* out = (float*)d_out;
  float* vp  = out;                    // 768*128
  float* sp  = out + NP * DF;          // 768*128
  float* ve2 = out + 2 * NP * DF;      // 768*768
  float* se2 = ve2 + NP * NP;          // 768*768

  float* ws     = (float*)d_ws;
  float* ve     = ws;                  // 768*768
  float* se     = ve + NP * NP;        // 768*768
  float* logits = se + NP * NP;        // 768*768 scratch
  float* pbuf   = logits + NP * NP;    // 768*128
  float* pebuf  = pbuf + NP * DF;      // 768*128
  float* tbuf   = pebuf + NP * DF;     // 768*128
  float* qbuf   = tbuf + NP * DF;      // 768*8
  float* nrmv   = qbuf + NP * HH;      // 768
  float* nrma   = nrmv + NP;           // 768

  dim3 tiles(NP / 16, NP / 16);        // 48x48
  dim3 gemmNTiles(DF / 16, NP / 16);   // 8x48
  dim3 w(32);

  // ---- edge init: ve = softmax(cos(VP)/0.1), se = softmax(cos(ATT)/0.1) ----
  rownorm_kernel<<<NP, 256, 0, stream>>>(VP, DF, nrmv);
  rownorm_kernel<<<NP, 256, 0, stream>>>(ATT, AD, nrma);
  gram_logits_kernel<<<tiles, w, 0, stream>>>(VP, nrmv, DF, 10.0f, logits);
  softmax768_kernel<<<NP, 256, 0, stream>>>(logits, ve);
  gram_logits_kernel<<<tiles, w, 0, stream>>>(ATT, nrma, AD, 10.0f, logits);
  softmax768_kernel<<<NP, 256, 0, stream>>>(logits, se);

  // ---- visual node update: p = ve@VP, extra = se@VP ----
  gemm_ab_kernel<<<gemmNTiles, w, 0, stream>>>(ve, VP, NP, NP, DF, pbuf, DF);
  gemm_ab_kernel<<<gemmNTiles, w, 0, stream>>>(se, VP, NP, NP, DF, pebuf, DF);
  gemm_xwT_bias_kernel<<<gemmNTiles, w, 0, stream>>>(pbuf, pebuf, Wvn, bvn, tbuf);
  ln_relu_add_kernel<<<NP, DF, 0, stream>>>(tbuf, pbuf, gvn, betavn, vp);

  // ---- visual edge update: ve2 = softmax(tanh(MLP(|vp_i - vp_j|^2)) * (ve+eps)/10) ----
  q_kernel<<<(NP * HH + 255) / 256, 256, 0, stream>>>(vp, Wve1, qbuf);
  edge_tile_kernel<<<tiles, w, 0, stream>>>(vp, qbuf, Wve1, bve1, Wve2, bve2, ve, 0.1f, logits);
  softmax768_kernel<<<NP, 256, 0, stream>>>(logits, ve2);

  // ---- semantic node update: p = se@SP, extra = ve2@SP ----
  gemm_ab_kernel<<<gemmNTiles, w, 0, stream>>>(se, SPm, NP, NP, DF, pbuf, DF);
  gemm_ab_kernel<<<gemmNTiles, w, 0, stream>>>(ve2, SPm, NP, NP, DF, pebuf, DF);
  gemm_xwT_bias_kernel<<<gemmNTiles, w, 0, stream>>>(pbuf, pebuf, Wsn, bsn, tbuf);
  ln_relu_add_kernel<<<NP, DF, 0, stream>>>(tbuf, pbuf, gsn, betasn, sp);

  // ---- semantic edge update ----
  q_kernel<<<(NP * HH + 255) / 256, 256, 0, stream>>>(sp, Wse1, qbuf);
  edge_tile_kernel<<<tiles, w, 0, stream>>>(sp, qbuf, Wse1, bse1, Wse2, bse2, se, 0.1f, logits);
  softmax768_kernel<<<NP, 256, 0, stream>>>(logits, se2);
}